// URNNCell_65893388255684
// MI455X (gfx1250) — compile-verified
//
#include <hip/hip_runtime.h>
#include <hip/hip_bf16.h>

// ---------------------------------------------------------------------------
// URNN cell for MI455X (gfx1250, wave32).
//  K1: fp32 WMMA GEMM  inputs(8192x256) @ w_ih^T(256x1024) -> ws
//      A-tile staged to LDS via async DMA (global_load_async_to_lds_b128).
//  K2: per-row complex chain: async-DMA state row -> diag -> FFT512 ->
//      reflect -> perm -> diag -> IFFT512 -> reflect -> diag -> +GEMM ->
//      modReLU -> out
// ---------------------------------------------------------------------------

#define BATCH     8192
#define NUM_IN    256
#define NUM_UNITS 512
#define PI_F      3.14159265358979323846f

typedef __attribute__((ext_vector_type(2))) float v2f;
typedef __attribute__((ext_vector_type(8))) float v8f;

// Async global->LDS copy of 16 bytes per lane. lds_off = LDS byte offset
// (low 32 bits of the generic shared pointer, per flat-aperture rule).
__device__ __forceinline__
void async_copy_b128(uint32_t lds_off, const void* gaddr)
{
    asm volatile("global_load_async_to_lds_b128 %0, %1, off"
                 :: "v"(lds_off), "v"(gaddr) : "memory");
}
__device__ __forceinline__ void async_wait_all()
{
    asm volatile("s_wait_asynccnt 0x0" ::: "memory");
}

// ---------------------------------------------------------------------------
// Kernel 1: C[m,n] = sum_k A[m,k] * W[n,k]
// Block: 256 threads = 8 waves. Tile: 16 (M) x 128 (N). Each wave: one 16x16.
// ---------------------------------------------------------------------------
#define A_STRIDE 260   // 260 % 64 = 4 -> 16 M-rows map to 16 distinct LDS banks

__global__ __launch_bounds__(256)
void urnn_gemm_wmma(const float* __restrict__ A,   // 8192 x 256
                    const float* __restrict__ W,   // 1024 x 256
                    float* __restrict__ C)         // 8192 x 1024
{
    __shared__ float As[16 * A_STRIDE];

    const int m0 = blockIdx.x * 16;
    const int n0 = blockIdx.y * 128;
    const int t  = threadIdx.x;
    const int wave = t >> 5;
    const int lane = t & 31;

    // Async DMA the 16x256 A tile into LDS: 1024 x b128, 4 per thread.
    // LDS addresses 16B aligned: row stride 260*4 = 65*16 B, col mult. of 16 B.
    {
        int i = t;
        #pragma unroll
        for (int it = 0; it < 4; ++it, i += 256) {
            const int r = i >> 6;              // row 0..15
            const int c = (i & 63) << 2;       // col 0..252 step 4
            const uint32_t loff =
                (uint32_t)(uintptr_t)(&As[r * A_STRIDE + c]);
            async_copy_b128(loff, A + (m0 + r) * NUM_IN + c);
        }
        async_wait_all();
        __syncthreads();
    }

    // Fragment coordinates per ISA 7.12.2 (32-bit A 16x4 / B 4x16 layouts):
    //   lanes 0-15 hold K pair {0,1}, lanes 16-31 hold K pair {2,3}.
    const int kk   = (lane >= 16) ? 2 : 0;
    const int mrow = lane & 15;                    // A: M = lane % 16
    const int ncol = n0 + wave * 16 + (lane & 15); // B: N = lane % 16
    const float* __restrict__ wp = W + ncol * NUM_IN;
    __builtin_prefetch(wp, 0, 3);                  // global_prefetch_b8

    v8f acc = {};
    #pragma unroll 4
    for (int k = 0; k < NUM_IN; k += 4) {
        v2f a, b;
        a.x = As[mrow * A_STRIDE + k + kk];
        a.y = As[mrow * A_STRIDE + k + kk + 1];
        b.x = wp[k + kk];
        b.y = wp[k + kk + 1];
        acc = __builtin_amdgcn_wmma_f32_16x16x4_f32(
                  false, a, false, b, (short)0, acc, false, false);
    }

    // C/D layout: VGPR i -> M = i (lanes 0-15) and M = i + 8 (lanes 16-31).
    const int mo = (lane >= 16) ? 8 : 0;
    #pragma unroll
    for (int i = 0; i < 8; ++i)
        C[(m0 + i + mo) * 1024 + ncol] = acc[i];
}

// ---------------------------------------------------------------------------
// Kernel 2 helpers. LDS arena layout (floats):
//   buffer b (b=0,1): re at lds[b*1024 + j], im at lds[b*1024 + 512 + j]
// ---------------------------------------------------------------------------

// Stockham radix-2 FFT, N=512, 256 threads, ping-pong buffers.
// sign = -1 forward, +1 inverse (caller applies 1/N for inverse).
__device__ __forceinline__
void fft512(float* lds, int& cur, int t, float sign)
{
    #pragma unroll
    for (int sh = 0; sh < 9; ++sh) {
        float* src = lds + cur * 1024;
        float* dst = lds + (cur ^ 1) * 1024;
        const int m  = 1 << sh;
        const int l  = 256 >> sh;
        const int jj = t >> sh;
        // Source indices collapse to t and t+256 (stride-1, conflict-free).
        const float ar = src[t],            ai = src[512 + t];
        const float cr = src[t + 256],      ci = src[512 + t + 256];
        float ws, wc;
        __sincosf(sign * PI_F * (float)jj / (float)l, &ws, &wc);
        const int w0 = (jj << (sh + 1)) | (t & (m - 1));
        dst[w0]       = ar + cr;
        dst[512 + w0] = ai + ci;
        const float dr = ar - cr, di = ai - ci;
        dst[w0 + m]       = dr * wc - di * ws;
        dst[512 + w0 + m] = dr * ws + di * wc;
        __syncthreads();
        cur ^= 1;
    }
}

// z -= (2/||v||^2) * (v^H z) * v   over buffer `buf`.
__device__ __forceinline__
void reflect(float* buf, float* red0, float* red1, float* red2, int t,
             const float* __restrict__ vre, const float* __restrict__ vim)
{
    float dr = 0.f, di = 0.f, nn = 0.f;
    #pragma unroll
    for (int j = t; j < NUM_UNITS; j += 256) {
        const float vr = vre[j], vi = vim[j];
        const float zr = buf[j], zi = buf[512 + j];
        dr += vr * zr + vi * zi;        // Re(conj(v) * z)
        di += vr * zi - vi * zr;        // Im(conj(v) * z)
        nn += vr * vr + vi * vi;
    }
    red0[t] = dr; red1[t] = di; red2[t] = nn;
    __syncthreads();
    for (int off = 128; off > 0; off >>= 1) {
        if (t < off) {
            red0[t] += red0[t + off];
            red1[t] += red1[t + off];
            red2[t] += red2[t + off];
        }
        __syncthreads();
    }
    const float s   = 2.0f / red2[0];
    const float sdr = red0[0] * s;
    const float sdi = red1[0] * s;
    __syncthreads();
    #pragma unroll
    for (int j = t; j < NUM_UNITS; j += 256) {
        const float vr = vre[j], vi = vim[j];
        buf[j]       -= sdr * vr - sdi * vi;
        buf[512 + j] -= sdr * vi + sdi * vr;
    }
    __syncthreads();
}

// ---------------------------------------------------------------------------
// Kernel 2: full unitary chain, one block (256 threads) per batch row.
// ---------------------------------------------------------------------------
__global__ __launch_bounds__(256)
void urnn_chain(const float* __restrict__ states,      // 8192 x 1024
                const float* __restrict__ inputs_mul,  // 8192 x 1024 (ws)
                const float* __restrict__ b_h,
                const float* __restrict__ d1,
                const float* __restrict__ r1re, const float* __restrict__ r1im,
                const long long* __restrict__ perm,
                const float* __restrict__ d2,
                const float* __restrict__ r2re, const float* __restrict__ r2im,
                const float* __restrict__ d3,
                float* __restrict__ out)               // 8192 x 1024
{
    __shared__ float lds[2 * 2 * NUM_UNITS];           // 8 KB ping-pong arena
    __shared__ float red0[256], red1[256], red2[256];

    const int t   = threadIdx.x;
    const int row = blockIdx.x;

    // Async DMA the raw interleaved state row (1024 f32) into buffer 1.
    {
        const uint32_t loff =
            (uint32_t)(uintptr_t)(lds + 1024 + t * 4);
        async_copy_b128(loff, states + row * (2 * NUM_UNITS) + t * 4);
        async_wait_all();
        __syncthreads();
    }

    // De-interleave + apply diag(e^{i d1}) into buffer 0.
    {
        const float* raw = lds + 1024;
        #pragma unroll
        for (int j = t; j < NUM_UNITS; j += 256) {
            const float re = raw[2 * j], im = raw[2 * j + 1];
            float s, c; __sincosf(d1[j], &s, &c);
            lds[j]       = c * re - s * im;
            lds[512 + j] = s * re + c * im;
        }
        __syncthreads();
    }

    int cur = 0;
    fft512(lds, cur, t, -1.0f);                        // forward FFT
    reflect(lds + cur * 1024, red0, red1, red2, t, r1re, r1im);

    // Permute columns and apply diag(e^{i d2}) into the other buffer.
    {
        float* src = lds + cur * 1024;
        float* dst = lds + (cur ^ 1) * 1024;
        #pragma unroll
        for (int j = t; j < NUM_UNITS; j += 256) {
            const int p = (int)perm[j];
            const float zr = src[p], zi = src[512 + p];
            float s, c; __sincosf(d2[j], &s, &c);
            dst[j]       = c * zr - s * zi;
            dst[512 + j] = s * zr + c * zi;
        }
        __syncthreads();
        cur ^= 1;
    }

    fft512(lds, cur, t, +1.0f);                        // inverse FFT...
    {
        float* buf = lds + cur * 1024;
        #pragma unroll
        for (int j = t; j < NUM_UNITS; j += 256) {     // ...with 1/N
            buf[j]       *= (1.0f / NUM_UNITS);
            buf[512 + j] *= (1.0f / NUM_UNITS);
        }
        __syncthreads();
    }

    reflect(lds + cur * 1024, red0, red1, red2, t, r2re, r2im);

    // diag(e^{i d3}) + GEMM result + modReLU, write concat(real, imag).
    const float* buf = lds + cur * 1024;
    const float* __restrict__ gm = inputs_mul + row * (2 * NUM_UNITS);
    float* __restrict__ orow = out + row * (2 * NUM_UNITS);
    #pragma unroll
    for (int j = t; j < NUM_UNITS; j += 256) {
        float s, c; __sincosf(d3[j], &s, &c);
        const float zr = buf[j], zi = buf[512 + j];
        const float pr = gm[2 * j]     + (c * zr - s * zi);
        const float pi = gm[2 * j + 1] + (s * zr + c * zi);
        const float norm = sqrtf(pr * pr + pi * pi);
        const float sc   = fmaxf(norm + b_h[j], 0.0f) / (norm + 1e-6f);
        orow[j]             = sc * pr;
        orow[NUM_UNITS + j] = sc * pi;
    }
}

// ---------------------------------------------------------------------------
// Launch
// ---------------------------------------------------------------------------
extern "C" void kernel_launch(void* const* d_in, const int* in_sizes, int n_in,
                              void* d_out, int out_size, void* d_ws, size_t ws_size,
                              hipStream_t stream)
{
    const float*     inputs = (const float*)    d_in[0];
    const float*     states = (const float*)    d_in[1];
    const float*     w_ih   = (const float*)    d_in[2];
    const float*     b_h    = (const float*)    d_in[3];
    const float*     d1     = (const float*)    d_in[4];
    const float*     r1re   = (const float*)    d_in[5];
    const float*     r1im   = (const float*)    d_in[6];
    const long long* perm   = (const long long*)d_in[7];
    const float*     d2     = (const float*)    d_in[8];
    const float*     r2re   = (const float*)    d_in[9];
    const float*     r2im   = (const float*)    d_in[10];
    const float*     d3     = (const float*)    d_in[11];
    float* out        = (float*)d_out;
    float* inputs_mul = (float*)d_ws;   // 8192*1024 f32 = 32 MB scratch

    dim3 g1(BATCH / 16, 1024 / 128);    // 512 x 8 blocks
    urnn_gemm_wmma<<<g1, 256, 0, stream>>>(inputs, w_ih, inputs_mul);

    urnn_chain<<<BATCH, 256, 0, stream>>>(states, inputs_mul, b_h, d1,
                                          r1re, r1im, perm, d2,
                                          r2re, r2im, d3, out);
}